// DigitCapsules_60455959658895
// MI455X (gfx1250) — compile-verified
//
#include <hip/hip_runtime.h>
#include <hip/hip_fp16.h>

typedef __attribute__((ext_vector_type(2))) float v2f;
typedef __attribute__((ext_vector_type(8))) float v8f;

#define Bsz 256
#define Isz 1152
#define Osz 10
#define Dsz 16
#define Ksz 8
#define ODsz 160   // O*D, od = o*16 + d

// ---------------------------------------------------------------------------
// Kernel 1: u_hat[b,i,od] = sum_k W[i,od,k] * x[b,i,k]   (stored fp16)
// One wave handles one i for a 16-batch tile: [16b x 8k] @ [8k x 160od]
// via 10 od-tiles, each = 2 chained V_WMMA_F32_16X16X4_F32.
// C-frag (lane=od-col, vgpr=batch-row) is transposed through wave-private
// LDS (row stride 162 dwords: 8*162 % 64 == 16 -> halves hit disjoint banks)
// then stored coalesced as __half2.
// ---------------------------------------------------------------------------
__global__ __launch_bounds__(128) void k_uhat(const float* __restrict__ x,
                                              const float* __restrict__ W,
                                              __half* __restrict__ uhat) {
  __shared__ float tr[4 * 16 * 162];           // 4 waves * 16 rows * 162 pad
  const int t    = threadIdx.x;
  const int wv   = t >> 5;
  const int lane = t & 31;
  const int m    = lane & 15;                  // batch row (A) / od col (B,C)
  const int kh   = lane >> 4;                  // which k-pair this lane holds
  const int i    = blockIdx.x * 4 + wv;        // 288 * 4 = 1152
  const int b0   = blockIdx.y * 16;            // 16 * 16  = 256

  // A-frag: x[b0+m, i, k];  a0 covers k=0..3, a1 covers k=4..7
  const float* xp = x + (((size_t)(b0 + m) * Isz + i) * Ksz) + 2 * kh;
  v2f a0 = { xp[0], xp[1] };
  v2f a1 = { xp[4], xp[5] };

  float* lw = tr + wv * (16 * 162);
  const float* wp = W + (((size_t)i * ODsz + m) * Ksz) + 2 * kh;  // n == m

#pragma unroll
  for (int o = 0; o < Osz; ++o) {
    const float* wo = wp + o * 16 * Ksz;       // B-frag: W[i, o*16+n, k]
    v2f bb0 = { wo[0], wo[1] };
    v2f bb1 = { wo[4], wo[5] };
    v8f acc = {};
    acc = __builtin_amdgcn_wmma_f32_16x16x4_f32(false, a0, false, bb0,
                                                (short)0, acc, false, false);
    acc = __builtin_amdgcn_wmma_f32_16x16x4_f32(false, a1, false, bb1,
                                                (short)0, acc, false, false);
    // C layout: lanes 0-15 vgpr r -> (M=r, N=lane); lanes 16-31 -> (M=8+r)
#pragma unroll
    for (int r = 0; r < 8; ++r)
      lw[(r + 8 * kh) * 162 + o * 16 + m] = acc[r];
  }

  // Wave-private transpose readout -> coalesced fp16 global stores.
  __half2* up = (__half2*)uhat;
#pragma unroll 1
  for (int mm = 0; mm < 16; ++mm) {
    const float* lr = lw + mm * 162;
    size_t gbase = ((size_t)(b0 + mm) * Isz + i) * (ODsz / 2);
#pragma unroll
    for (int c = 0; c < 3; ++c) {
      int od2 = c * 32 + lane;                 // 80 half2 per (b,i) row
      if (od2 < 80)
        up[gbase + od2] = __floats2half2_rn(lr[od2 * 2], lr[od2 * 2 + 1]);
    }
  }
}

// ---------------------------------------------------------------------------
// Kernel 2 (fused routing step): for one (b, 128-i chunk):
//   stage u_hat chunk in LDS (one HBM pass),
//   [r>0]  logits += sum_d u_hat * v_prev   (write back for next iter),
//   c = softmax_o(logits), partial s[b,od] += sum_i c * u_hat  (f32 atomics)
// ---------------------------------------------------------------------------
__global__ __launch_bounds__(320) void k_route(const __half* __restrict__ uhat,
                                               float* __restrict__ logits,
                                               const float* __restrict__ vprev,
                                               float* __restrict__ sout,
                                               int update) {
  __shared__ __half uh_s[128 * ODsz];          // 40 KB
  __shared__ float  c_s[128 * Osz];            // logits -> c, in place
  __shared__ float  v_s[ODsz];
  const int t  = threadIdx.x;
  const int b  = blockIdx.y;
  const int i0 = blockIdx.x * 128;

  // stage 40KB of u_hat (contiguous, 16B vectors)
  const uint4* src = (const uint4*)(uhat + ((size_t)b * Isz + i0) * ODsz);
  uint4* dst = (uint4*)uh_s;
  for (int j = t; j < 2560; j += 320) dst[j] = src[j];
  if (update && t < ODsz) v_s[t] = vprev[b * ODsz + t];
  __syncthreads();

  if (update) {
    for (int p = t; p < 1280; p += 320) {      // p = i*10 + o
      int i = p / 10, o = p - 10 * i;
      const __half* u = uh_s + i * ODsz + o * 16;
      float acc = 0.f;
#pragma unroll
      for (int d = 0; d < 16; ++d) acc += (float)u[d] * v_s[o * 16 + d];
      size_t gi = ((size_t)b * Isz + i0 + i) * Osz + o;
      float lg = logits[gi] + acc;
      logits[gi] = lg;
      c_s[p] = lg;
    }
  } else {
    for (int p = t; p < 1280; p += 320) c_s[p] = 0.f;   // softmax(0) = 0.1
  }
  __syncthreads();

  if (t < 128) {                               // softmax over O=10 per row
    float mx = c_s[t * 10];
#pragma unroll
    for (int o = 1; o < 10; ++o) mx = fmaxf(mx, c_s[t * 10 + o]);
    float e[10], sum = 0.f;
#pragma unroll
    for (int o = 0; o < 10; ++o) { e[o] = __expf(c_s[t * 10 + o] - mx); sum += e[o]; }
    float inv = 1.f / sum;
#pragma unroll
    for (int o = 0; o < 10; ++o) c_s[t * 10 + o] = e[o] * inv;
  }
  __syncthreads();

  // weighted sum over this i-chunk: 320 threads = 160 od * 2 halves
  const int od = t % ODsz, hf = t / ODsz, o = od >> 4;
  float acc = 0.f;
  for (int il = hf * 64; il < hf * 64 + 64; ++il)
    acc += c_s[il * 10 + o] * (float)uh_s[il * ODsz + od];
  atomicAdd(&sout[b * ODsz + od], acc);
}

// ---------------------------------------------------------------------------
// Kernel 3: v = squash(s) per (b,o) vector of 16
// ---------------------------------------------------------------------------
__global__ __launch_bounds__(256) void k_squash(const float* __restrict__ s,
                                                float* __restrict__ v) {
  int id = blockIdx.x * 256 + threadIdx.x;
  if (id >= Bsz * Osz) return;
  const float* sp = s + id * Dsz;
  float vals[16], sq = 0.f;
#pragma unroll
  for (int d = 0; d < 16; ++d) { vals[d] = sp[d]; sq += vals[d] * vals[d]; }
  float scale = (sq / (1.f + sq)) / sqrtf(sq + 1e-8f);
  float* vp = v + id * Dsz;
#pragma unroll
  for (int d = 0; d < 16; ++d) vp[d] = vals[d] * scale;
}

// ---------------------------------------------------------------------------
extern "C" void kernel_launch(void* const* d_in, const int* in_sizes, int n_in,
                              void* d_out, int out_size, void* d_ws, size_t ws_size,
                              hipStream_t stream) {
  const float* x = (const float*)d_in[0];   // [256,1152,8]
  const float* W = (const float*)d_in[1];   // [1152,10,16,8]
  float* out = (float*)d_out;               // [256,10,16]
  char* ws = (char*)d_ws;

  const size_t UHAT_B   = (size_t)Bsz * Isz * ODsz * 2;   // 94,371,840
  const size_t LOGITS_B = (size_t)Bsz * Isz * Osz * 4;    // 11,796,480
  const size_t SV_ELEMS = (size_t)Bsz * ODsz;             // 40,960 floats

  __half* uhat  = (__half*)ws;
  float* logits = (float*)(ws + UHAT_B);
  float* sbuf   = (float*)(ws + UHAT_B + LOGITS_B);       // 3 * SV
  float* vbuf   = (float*)(ws + UHAT_B + LOGITS_B + 3 * SV_ELEMS * 4); // 2 * SV

  // logits and the 3 s-buffers are contiguous: one async memset (capture-safe)
  hipMemsetAsync(logits, 0, LOGITS_B + 3 * SV_ELEMS * 4, stream);

  k_uhat<<<dim3(Isz / 4, Bsz / 16), 128, 0, stream>>>(x, W, uhat);

  for (int r = 0; r < 3; ++r) {
    float* s_r = sbuf + (size_t)r * SV_ELEMS;
    const float* vp = r ? (vbuf + (size_t)(r - 1) * SV_ELEMS) : nullptr;
    k_route<<<dim3(Isz / 128, Bsz), 320, 0, stream>>>(uhat, logits, vp, s_r, r);
    float* vout = (r == 2) ? out : (vbuf + (size_t)r * SV_ELEMS);
    k_squash<<<dim3(10), 256, 0, stream>>>(s_r, vout);
  }
}